// VisionTransformer_37538014167350
// MI455X (gfx1250) — compile-verified
//
#include <hip/hip_runtime.h>

typedef __attribute__((ext_vector_type(2))) float v2f;
typedef __attribute__((ext_vector_type(8))) float v8f;

#define AS3 __attribute__((address_space(3)))
typedef AS3 float lfloat;
typedef AS3 v2f   lv2f;

#define SH 258   // stride of h / t planes (floats)
#define SQ 34    // stride of Qh / Vh planes
#define SK 65    // stride of K-transposed plane
#define SS 66    // stride of score plane
#define HID 256
#define SEQ 64

// union region U sizes (floats)
#define USIZE 8512                  // max( Q(2*64*34)+KT(2*32*65), SC(2*64*66) )
#define SMEM_FLOATS (4*64*SH + USIZE + 2*64*SQ + 1024)

static __device__ __forceinline__ v8f vzero() {
    v8f z = {0.f,0.f,0.f,0.f,0.f,0.f,0.f,0.f};
    return z;
}

// D = A(16x4 f32) * B(4x16 f32) + C  — CDNA5 fp32 WMMA
static __device__ __forceinline__ v8f wmma4(v2f a, v2f b, v8f c) {
    return __builtin_amdgcn_wmma_f32_16x16x4_f32(
        false, a, false, b, (short)0, c, false, false);
}

// complex tanh-approx gelu (matches jnp gelu on complex)
static __device__ __forceinline__ void cgelu(float& zr, float& zi) {
    const float c0 = 0.7978845608028654f;   // sqrt(2/pi)
    float z2r = zr*zr - zi*zi,  z2i = 2.f*zr*zi;
    float z3r = z2r*zr - z2i*zi, z3i = z2r*zi + z2i*zr;
    float ur = c0*(zr + 0.044715f*z3r);
    float ui = c0*(zi + 0.044715f*z3i);
    float th = tanhf(ur), tn = tanf(ui);
    // tanh(u) = (th + i*tn) / (1 + i*th*tn)
    float dr = 1.f, di = th*tn;
    float inv = 1.f/(dr*dr + di*di);
    float tr = (th*dr + tn*di)*inv;
    float ti = (tn*dr - th*di)*inv;
    float opr = 1.f + tr;
    float orr = 0.5f*(zr*opr - zi*ti);
    float oii = 0.5f*(zr*ti + zi*opr);
    zr = orr; zi = oii;
}

// out[64,32] = h[64,256] @ W[:, wcol0:wcol0+32] + bias   (complex)
// transpose==1: store as out[d][s] (for K^T)
static __device__ void gemm_qkv(const lfloat* Ar, const lfloat* Ai,
                                const float2* __restrict__ W, int wcol0,
                                const float2* __restrict__ bias,
                                lfloat* Or, lfloat* Oi,
                                int ldo, int transpose, int tid)
{
    const int wave = tid >> 5, lane = tid & 31, subl = lane & 15, hl = lane >> 4;
    const int rt = wave & 3, ct = wave >> 2;
    const int colW = wcol0 + ct*16 + subl;
    v8f accr = vzero(), acci = vzero();
#pragma unroll 1
    for (int k = 0; k < HID; k += 4) {
        const int arow = (rt*16 + subl)*SH + k + 2*hl;
        v2f ar = *(const lv2f*)(Ar + arow);
        v2f ai = *(const lv2f*)(Ai + arow);
        v2f an = -ai;
        const int kk = k + 2*hl;
        float2 w0 = W[(size_t)kk*HID + colW];
        float2 w1 = W[(size_t)(kk+1)*HID + colW];
        v2f br; br.x = w0.x; br.y = w1.x;
        v2f bi; bi.x = w0.y; bi.y = w1.y;
        accr = wmma4(an, bi, accr);
        accr = wmma4(ar, br, accr);
        acci = wmma4(ar, bi, acci);
        acci = wmma4(ai, br, acci);
    }
    float2 bv = bias[colW];
#pragma unroll
    for (int r = 0; r < 8; ++r) {
        int row = rt*16 + r + 8*hl;
        float zr = accr[r] + bv.x;
        float zi = acci[r] + bv.y;
        if (!transpose) {
            Or[row*ldo + ct*16 + subl] = zr;
            Oi[row*ldo + ct*16 + subl] = zi;
        } else {
            Or[(ct*16 + subl)*ldo + row] = zr;
            Oi[(ct*16 + subl)*ldo + row] = zi;
        }
    }
}

// scores[64,64] = (Q @ K^T) * 1/sqrt(32)   (complex), written into union (aliases Q/K^T)
static __device__ void scores_gemm(const lfloat* qR, const lfloat* qI,
                                   const lfloat* kTR, const lfloat* kTI,
                                   lfloat* scR, lfloat* scI, int tid)
{
    const int wave = tid >> 5, lane = tid & 31, subl = lane & 15, hl = lane >> 4;
    const int rt = wave >> 1, ct0 = (wave & 1)*2;
    v8f accr[2], acci[2];
    accr[0]=vzero(); accr[1]=vzero(); acci[0]=vzero(); acci[1]=vzero();
#pragma unroll 1
    for (int k = 0; k < 32; k += 4) {
        const int arow = (rt*16 + subl)*SQ + k + 2*hl;
        v2f ar = *(const lv2f*)(qR + arow);
        v2f ai = *(const lv2f*)(qI + arow);
        v2f an = -ai;
        const int kk = k + 2*hl;
#pragma unroll
        for (int ct = 0; ct < 2; ++ct) {
            int col = (ct0+ct)*16 + subl;
            v2f br; br.x = kTR[kk*SK + col]; br.y = kTR[(kk+1)*SK + col];
            v2f bi; bi.x = kTI[kk*SK + col]; bi.y = kTI[(kk+1)*SK + col];
            accr[ct] = wmma4(an, bi, accr[ct]);
            accr[ct] = wmma4(ar, br, accr[ct]);
            acci[ct] = wmma4(ar, bi, acci[ct]);
            acci[ct] = wmma4(ai, br, acci[ct]);
        }
    }
    const float sc = 0.17677669529663687f;  // 1/sqrt(32)
    __syncthreads();   // all waves done reading Q/K^T before aliasing writes
#pragma unroll
    for (int ct = 0; ct < 2; ++ct)
#pragma unroll
        for (int r = 0; r < 8; ++r) {
            int row = rt*16 + r + 8*hl;
            int col = (ct0+ct)*16 + subl;
            scR[row*SS + col] = accr[ct][r]*sc;
            scI[row*SS + col] = acci[ct][r]*sc;
        }
    __syncthreads();
}

// complex softmax over rows of sc[64,64], in place
static __device__ void softmax64(lfloat* scR, lfloat* scI, lfloat* red, int tid)
{
    const int row = tid >> 2, tg = tid & 3;
    float m = -1e30f;
#pragma unroll
    for (int j = 0; j < 16; ++j) m = fmaxf(m, scR[row*SS + tg*16 + j]);
    red[tid] = m;
    __syncthreads();
    const int base = row*4;
    m = fmaxf(fmaxf(red[base], red[base+1]), fmaxf(red[base+2], red[base+3]));
    float sr = 0.f, si = 0.f;
#pragma unroll
    for (int j = 0; j < 16; ++j) {
        int c = row*SS + tg*16 + j;
        float er = expf(scR[c] - m);
        float sy, cy; sincosf(scI[c], &sy, &cy);
        float pr = er*cy, pi = er*sy;
        scR[c] = pr; scI[c] = pi;
        sr += pr; si += pi;
    }
    red[256 + tid] = sr; red[512 + tid] = si;
    __syncthreads();
    float Sr = red[256+base] + red[256+base+1] + red[256+base+2] + red[256+base+3];
    float Si = red[512+base] + red[512+base+1] + red[512+base+2] + red[512+base+3];
    float inv = 1.f/(Sr*Sr + Si*Si);
#pragma unroll
    for (int j = 0; j < 16; ++j) {
        int c = row*SS + tg*16 + j;
        float pr = scR[c], pi = scI[c];
        scR[c] = (pr*Sr + pi*Si)*inv;
        scI[c] = (pi*Sr - pr*Si)*inv;
    }
    __syncthreads();
}

// t[:, colOff:colOff+32] = P[64,64] @ V[64,32]   (complex)
static __device__ void pv_gemm(const lfloat* scR, const lfloat* scI,
                               const lfloat* vRp, const lfloat* vIp,
                               lfloat* tR, lfloat* tI,
                               int colOff, int tid)
{
    const int wave = tid >> 5, lane = tid & 31, subl = lane & 15, hl = lane >> 4;
    const int rt = wave & 3, ct = wave >> 2;
    v8f accr = vzero(), acci = vzero();
#pragma unroll 1
    for (int k = 0; k < SEQ; k += 4) {
        const int arow = (rt*16 + subl)*SS + k + 2*hl;
        v2f ar = *(const lv2f*)(scR + arow);
        v2f ai = *(const lv2f*)(scI + arow);
        v2f an = -ai;
        const int kk = k + 2*hl;
        int col = ct*16 + subl;
        v2f br; br.x = vRp[kk*SQ + col]; br.y = vRp[(kk+1)*SQ + col];
        v2f bi; bi.x = vIp[kk*SQ + col]; bi.y = vIp[(kk+1)*SQ + col];
        accr = wmma4(an, bi, accr);
        accr = wmma4(ar, br, accr);
        acci = wmma4(ar, bi, acci);
        acci = wmma4(ai, br, acci);
    }
#pragma unroll
    for (int r = 0; r < 8; ++r) {
        int row = rt*16 + r + 8*hl;
        int col = colOff + ct*16 + subl;
        tR[row*SH + col] = accr[r];
        tI[row*SH + col] = acci[r];
    }
}

// out[64,256] = A[64,256] @ W[256,256] + bias   (complex), optional complex-gelu
static __device__ void gemm256(const lfloat* Ar, const lfloat* Ai,
                               const float2* __restrict__ W, const float2* __restrict__ bias,
                               lfloat* Or, lfloat* Oi,
                               int dogelu, int tid)
{
    const int wave = tid >> 5, lane = tid & 31, subl = lane & 15, hl = lane >> 4;
    const int ct0 = wave*2;
    v8f accr[4][2], acci[4][2];
#pragma unroll
    for (int rt = 0; rt < 4; ++rt)
#pragma unroll
        for (int ct = 0; ct < 2; ++ct) { accr[rt][ct] = vzero(); acci[rt][ct] = vzero(); }
#pragma unroll 1
    for (int k = 0; k < HID; k += 4) {
        v2f ar[4], ai[4], an[4];
#pragma unroll
        for (int rt = 0; rt < 4; ++rt) {
            const int arow = (rt*16 + subl)*SH + k + 2*hl;
            ar[rt] = *(const lv2f*)(Ar + arow);
            ai[rt] = *(const lv2f*)(Ai + arow);
            an[rt] = -ai[rt];
        }
        const int kk = k + 2*hl;
        v2f br[2], bi[2];
#pragma unroll
        for (int ct = 0; ct < 2; ++ct) {
            int col = (ct0+ct)*16 + subl;
            float2 w0 = W[(size_t)kk*HID + col];
            float2 w1 = W[(size_t)(kk+1)*HID + col];
            br[ct].x = w0.x; br[ct].y = w1.x;
            bi[ct].x = w0.y; bi[ct].y = w1.y;
        }
#pragma unroll
        for (int rt = 0; rt < 4; ++rt)
#pragma unroll
            for (int ct = 0; ct < 2; ++ct) {
                accr[rt][ct] = wmma4(an[rt], bi[ct], accr[rt][ct]);
                accr[rt][ct] = wmma4(ar[rt], br[ct], accr[rt][ct]);
                acci[rt][ct] = wmma4(ar[rt], bi[ct], acci[rt][ct]);
                acci[rt][ct] = wmma4(ai[rt], br[ct], acci[rt][ct]);
            }
    }
#pragma unroll
    for (int ct = 0; ct < 2; ++ct) {
        int col = (ct0+ct)*16 + subl;
        float2 bv = bias[col];
#pragma unroll
        for (int rt = 0; rt < 4; ++rt)
#pragma unroll
            for (int r = 0; r < 8; ++r) {
                int row = rt*16 + r + 8*hl;
                float zr = accr[rt][ct][r] + bv.x;
                float zi = acci[rt][ct][r] + bv.y;
                if (dogelu) cgelu(zr, zi);
                Or[row*SH + col] = zr;
                Oi[row*SH + col] = zi;
            }
    }
}

// complex LayerNorm over last dim (256); useAdd: add residual from addR/addI
static __device__ void layernorm(lfloat* hR, lfloat* hI,
                                 const lfloat* addR, const lfloat* addI, int useAdd,
                                 const float2* __restrict__ sc, const float2* __restrict__ bc,
                                 lfloat* red, int tid)
{
    const int row = tid >> 2, tg = tid & 3;
    float sr = 0.f, si = 0.f, sq = 0.f;
    for (int j = 0; j < 64; ++j) {
        int c = tg + 4*j;
        float zr = hR[row*SH + c], zi = hI[row*SH + c];
        if (useAdd) { zr += addR[row*SH + c]; zi += addI[row*SH + c]; }
        sr += zr; si += zi; sq += zr*zr + zi*zi;
    }
    red[tid] = sr; red[256+tid] = si; red[512+tid] = sq;
    __syncthreads();
    const int base = row*4;
    float tsr = red[base]+red[base+1]+red[base+2]+red[base+3];
    float tsi = red[256+base]+red[256+base+1]+red[256+base+2]+red[256+base+3];
    float tsq = red[512+base]+red[512+base+1]+red[512+base+2]+red[512+base+3];
    float mur = tsr*(1.f/256.f), mui = tsi*(1.f/256.f);
    float var = tsq*(1.f/256.f) - (mur*mur + mui*mui);
    float inv = rsqrtf(var + 1e-6f);
    for (int j = 0; j < 64; ++j) {
        int c = tg + 4*j;
        float zr = hR[row*SH + c], zi = hI[row*SH + c];
        if (useAdd) { zr += addR[row*SH + c]; zi += addI[row*SH + c]; }
        float yr = (zr - mur)*inv, yi = (zi - mui)*inv;
        float2 s2 = sc[c]; float2 b2 = bc[c];
        hR[row*SH + c] = yr*s2.x - yi*s2.y + b2.x;
        hI[row*SH + c] = yr*s2.y + yi*s2.x + b2.y;
    }
    __syncthreads();
}

__global__ void __launch_bounds__(256)
cvit1250_kernel(const float* __restrict__ x,
                const float2* __restrict__ patch_w, const float2* __restrict__ patch_b,
                const float2* __restrict__ pos_emb,
                const float2* __restrict__ wq, const float2* __restrict__ bq,
                const float2* __restrict__ wk, const float2* __restrict__ bk,
                const float2* __restrict__ wv, const float2* __restrict__ bv,
                const float2* __restrict__ ln1_s, const float2* __restrict__ ln1_b,
                const float2* __restrict__ w1, const float2* __restrict__ b1,
                const float2* __restrict__ w2, const float2* __restrict__ b2,
                const float2* __restrict__ ln2_s, const float2* __restrict__ ln2_b,
                const float2* __restrict__ head_w, const float2* __restrict__ head_b,
                float* __restrict__ out)
{
    extern __shared__ float smem_raw[];
    lfloat* sm = (lfloat*)smem_raw;      // single explicit generic->LDS cast

    lfloat* hR  = sm;
    lfloat* hI  = hR + 64*SH;
    lfloat* tR  = hI + 64*SH;
    lfloat* tI  = tR + 64*SH;
    lfloat* U   = tI + 64*SH;
    lfloat* qR  = U;             lfloat* qI  = qR + 64*SQ;
    lfloat* kTR = qI + 64*SQ;    lfloat* kTI = kTR + 32*SK;
    lfloat* scR = U;             lfloat* scI = scR + 64*SS;   // aliases Q/K^T
    lfloat* vR  = U + USIZE;     lfloat* vI  = vR + 64*SQ;
    lfloat* red = vI + 64*SQ;                                  // 1024 floats

    const int tid = threadIdx.x;
    const int b = blockIdx.x;

    // ---------------- patch embed + pos emb ----------------
    {
        lfloat* xbuf = U;  // 1024 floats, reused before first layer
        for (int i = tid; i < 1024; i += 256) {
            int s = i >> 4, j = i & 15;
            int ph = s >> 3, pwc = s & 7;
            int p1 = j >> 2, p2 = j & 3;
            xbuf[i] = x[((size_t)b*32 + ph*4 + p1)*32 + (pwc*4 + p2)];
        }
        __syncthreads();
        float2 pwr[16];
#pragma unroll
        for (int j = 0; j < 16; ++j) pwr[j] = patch_w[j*HID + tid];
        float2 pb = patch_b[tid];
#pragma unroll 1
        for (int s = 0; s < SEQ; ++s) {
            float ar = pb.x, ai = pb.y;
#pragma unroll
            for (int j = 0; j < 16; ++j) {
                float xv = xbuf[s*16 + j];
                ar += xv*pwr[j].x; ai += xv*pwr[j].y;
            }
            float2 pe = pos_emb[s*HID + tid];
            hR[s*SH + tid] = ar + pe.x;
            hI[s*SH + tid] = ai + pe.y;
        }
        __syncthreads();
    }

    // ---------------- transformer layers ----------------
#pragma unroll 1
    for (int l = 0; l < 8; ++l) {
        const float2* Wq = wq + (size_t)l*HID*HID; const float2* Bq = bq + l*HID;
        const float2* Wk = wk + (size_t)l*HID*HID; const float2* Bk = bk + l*HID;
        const float2* Wv = wv + (size_t)l*HID*HID; const float2* Bv = bv + l*HID;
        const float2* W1 = w1 + (size_t)l*HID*HID; const float2* B1 = b1 + l*HID;
        const float2* W2 = w2 + (size_t)l*HID*HID; const float2* B2 = b2 + l*HID;

#pragma unroll 1
        for (int hd = 0; hd < 8; ++hd) {
            gemm_qkv(hR, hI, Wq, hd*32, Bq, qR,  qI,  SQ, 0, tid);
            gemm_qkv(hR, hI, Wk, hd*32, Bk, kTR, kTI, SK, 1, tid);
            gemm_qkv(hR, hI, Wv, hd*32, Bv, vR,  vI,  SQ, 0, tid);
            __syncthreads();
            scores_gemm(qR, qI, kTR, kTI, scR, scI, tid);
            softmax64(scR, scI, red, tid);
            pv_gemm(scR, scI, vR, vI, tR, tI, hd*32, tid);
            __syncthreads();
        }
        // h = LN1(h + attn)
        layernorm(hR, hI, tR, tI, 1, ln1_s + l*HID, ln1_b + l*HID, red, tid);
        // t = gelu(h @ w1 + b1)
        gemm256(hR, hI, W1, B1, tR, tI, 1, tid);
        __syncthreads();
        // h = t @ w2 + b2
        gemm256(tR, tI, W2, B2, hR, hI, 0, tid);
        __syncthreads();
        // h = LN2(h)
        layernorm(hR, hI, hR, hI, 0, ln2_s + l*HID, ln2_b + l*HID, red, tid);
    }

    // ---------------- log-cosh, mean over seq, head projection ----------------
    {
        float ar = 0.f, ai = 0.f;
#pragma unroll 1
        for (int s = 0; s < SEQ; ++s) {
            float zr = hR[s*SH + tid], zi = hI[s*SH + tid];
            float ex = expf(zr), emx = 1.f/ex;
            float ch = 0.5f*(ex + emx), sh = 0.5f*(ex - emx);
            float sy, cy; sincosf(zi, &sy, &cy);
            float wr = ch*cy, wi = sh*sy;                 // cosh(z)
            ar += 0.5f*logf(wr*wr + wi*wi);               // Re log cosh z
            ai += atan2f(wi, wr);                         // Im log cosh z
        }
        ar *= (1.f/64.f); ai *= (1.f/64.f);
        float2 hw = head_w[tid];
        float pr = ar*hw.x - ai*hw.y;
        float pi = ar*hw.y + ai*hw.x;
        red[tid] = pr; red[256 + tid] = pi;
        __syncthreads();
        for (int off = 128; off > 0; off >>= 1) {
            if (tid < off) {
                red[tid]       += red[tid + off];
                red[256 + tid] += red[256 + tid + off];
            }
            __syncthreads();
        }
        if (tid == 0) {
            float2 hb = head_b[0];
            out[2*b + 0] = red[0]   + hb.x;
            out[2*b + 1] = red[256] + hb.y;
        }
    }
}

extern "C" void kernel_launch(void* const* d_in, const int* in_sizes, int n_in,
                              void* d_out, int out_size, void* d_ws, size_t ws_size,
                              hipStream_t stream) {
    const float*  x       = (const float*) d_in[0];
    const float2* patch_w = (const float2*)d_in[1];
    const float2* patch_b = (const float2*)d_in[2];
    const float2* pos_emb = (const float2*)d_in[3];
    const float2* wq      = (const float2*)d_in[4];
    const float2* bq      = (const float2*)d_in[5];
    const float2* wk      = (const float2*)d_in[6];
    const float2* bk      = (const float2*)d_in[7];
    const float2* wv      = (const float2*)d_in[8];
    const float2* bv      = (const float2*)d_in[9];
    const float2* ln1_s   = (const float2*)d_in[10];
    const float2* ln1_b   = (const float2*)d_in[11];
    const float2* w1      = (const float2*)d_in[12];
    const float2* b1      = (const float2*)d_in[13];
    const float2* w2      = (const float2*)d_in[14];
    const float2* b2      = (const float2*)d_in[15];
    const float2* ln2_s   = (const float2*)d_in[16];
    const float2* ln2_b   = (const float2*)d_in[17];
    const float2* head_w  = (const float2*)d_in[18];
    const float2* head_b  = (const float2*)d_in[19];
    float* out = (float*)d_out;

    const int Bn = in_sizes[0] / (32*32);   // 512 batch items
    const size_t smem = (size_t)SMEM_FLOATS * sizeof(float);  // ~312 KB of the 320 KB WGP LDS
    hipFuncSetAttribute((const void*)cvit1250_kernel,
                        hipFuncAttributeMaxDynamicSharedMemorySize, (int)smem);
    cvit1250_kernel<<<Bn, 256, smem, stream>>>(
        x, patch_w, patch_b, pos_emb, wq, bq, wk, bk, wv, bv,
        ln1_s, ln1_b, w1, b1, w2, b2, ln2_s, ln2_b, head_w, head_b, out);
}